// SemiSupervisedGraphConvLayer_43499428774646
// MI455X (gfx1250) — compile-verified
//
#include <hip/hip_runtime.h>

typedef __attribute__((ext_vector_type(2))) float v2f;
typedef __attribute__((ext_vector_type(8))) float v8f;

#define NNODES 100000
#define NEDGES 1000000
#define DFEAT  64
#define NTILES (NNODES / 16)   // 6250, exact

// ---------------------------------------------------------------------------
// Kernel 1: agg = features  (residual folded into the scatter target)
// ---------------------------------------------------------------------------
__global__ __launch_bounds__(256) void init_agg(const float4* __restrict__ feat,
                                                float4* __restrict__ agg, int n4) {
    int i = blockIdx.x * blockDim.x + threadIdx.x;
    if (i < n4) agg[i] = feat[i];
}

// ---------------------------------------------------------------------------
// Kernel 2: wave-per-edge scatter-add.
// Lanes 0..31 each own a float2 slice of the 64-wide feature row, so the
// edge-index / edge-weight loads are wave-uniform (scalarized by the
// compiler) and both the gather and the atomics are coalesced 256B ops.
// features + agg live in L2 (25.6 MB each << 192 MB L2).
// ---------------------------------------------------------------------------
__global__ __launch_bounds__(256) void scatter_edges(const float* __restrict__ feat,
                                                     const long long* __restrict__ ei,
                                                     const float* __restrict__ ew,
                                                     float* __restrict__ agg) {
    int edge = blockIdx.x * (blockDim.x >> 5) + (threadIdx.x >> 5);
    int lane = threadIdx.x & 31;
    if (edge >= NEDGES) return;

    int   src = (int)ei[edge];           // wave-uniform
    int   dst = (int)ei[NEDGES + edge];  // wave-uniform
    float w   = ew[edge];                // wave-uniform

    const float2* srow = (const float2*)(feat + src * DFEAT);
    float2 m = srow[lane];

    float* dp = agg + dst * DFEAT + lane * 2;
    atomicAdd(dp,     m.x * w);
    atomicAdd(dp + 1, m.y * w);
}

// ---------------------------------------------------------------------------
// Kernel 3: OUT[100000,64] = X @ W^T + b via V_WMMA_F32_16X16X4_F32.
// Each wave computes a 16x64 output strip: 4 column tiles x 16 K-steps.
//
// VGPR layouts per CDNA5 ISA 7.12.2:
//  A (16x4 f32):  lane<16 -> M=lane,   {VGPR0,VGPR1} = K{0,1}
//                 lane>=16 -> M=lane-16,{VGPR0,VGPR1} = K{2,3}
//  B (4x16 f32):  lane<16 -> N=lane,   {VGPR0,VGPR1} = K{0,1}? (rows split as
//                 K=r / K=r+2 across lane halves, mirroring the C/D split)
//  C/D (16x16):   VGPR r: lanes 0-15 -> (M=r, N=lane); lanes 16-31 -> (M=r+8)
// B[k][n] = W[n0+n][k0+k]  (since OUT = X * W^T), which makes the B load the
// same float2 addressing shape as the A load.
// ---------------------------------------------------------------------------
__global__ __launch_bounds__(256) void gemm_wmma(const float* __restrict__ X,
                                                 const float* __restrict__ Wm,
                                                 const float* __restrict__ bias,
                                                 float* __restrict__ out) {
    int tile = blockIdx.x * (blockDim.x >> 5) + (threadIdx.x >> 5);
    if (tile >= NTILES) return;            // wave-uniform: EXEC stays all-ones

    int lane = threadIdx.x & 31;
    int half = lane >> 4;                  // 0: K pair {0,1}; 1: K pair {2,3}
    int l16  = lane & 15;
    int m0   = tile * 16;

    const float* arow = X + (m0 + l16) * DFEAT + 2 * half;

    v8f acc[4] = {};
    #pragma unroll 4
    for (int k0 = 0; k0 < DFEAT; k0 += 4) {
        v2f a = *(const v2f*)(arow + k0);
        #pragma unroll
        for (int t = 0; t < 4; ++t) {
            const float* brow = Wm + (t * 16 + l16) * DFEAT + k0 + 2 * half;
            v2f bb = *(const v2f*)brow;
            acc[t] = __builtin_amdgcn_wmma_f32_16x16x4_f32(
                false, a, false, bb, (short)0, acc[t], false, false);
        }
    }

    int row_off = half * 8;
    #pragma unroll
    for (int t = 0; t < 4; ++t) {
        int n = t * 16 + l16;
        float bv = bias[n];
        #pragma unroll
        for (int r = 0; r < 8; ++r)
            out[(m0 + row_off + r) * DFEAT + n] = acc[t][r] + bv;
    }
}

// ---------------------------------------------------------------------------
extern "C" void kernel_launch(void* const* d_in, const int* in_sizes, int n_in,
                              void* d_out, int out_size, void* d_ws, size_t ws_size,
                              hipStream_t stream) {
    const float*     feat = (const float*)d_in[0];      // [100000, 64] f32
    const long long* ei   = (const long long*)d_in[1];  // [2, 1000000] i64
    const float*     ew   = (const float*)d_in[2];      // [1000000] f32
    const float*     Wm   = (const float*)d_in[3];      // [64, 64] f32
    const float*     bias = (const float*)d_in[4];      // [64] f32
    float*           out  = (float*)d_out;              // [100000, 64] f32
    float*           agg  = (float*)d_ws;               // 25.6 MB scratch

    // 1) agg = features
    int n4 = NNODES * DFEAT / 4;
    init_agg<<<(n4 + 255) / 256, 256, 0, stream>>>((const float4*)feat,
                                                   (float4*)agg, n4);

    // 2) scatter-add edge messages (8 waves/block, wave per edge)
    int edge_blocks = (NEDGES + 7) / 8;
    scatter_edges<<<edge_blocks, 256, 0, stream>>>(feat, ei, ew, agg);

    // 3) (agg) @ W^T + b  — 8 wave-tiles (16 rows each) per block
    int gemm_blocks = (NTILES + 7) / 8;
    gemm_wmma<<<gemm_blocks, 256, 0, stream>>>(agg, Wm, bias, out);
}